// ShipGNN_23184233464436
// MI455X (gfx1250) — compile-verified
//
#include <hip/hip_runtime.h>

typedef __attribute__((ext_vector_type(2))) float        v2f;
typedef __attribute__((ext_vector_type(8))) float        v8f;
typedef __attribute__((ext_vector_type(4))) unsigned int v4u;
typedef __attribute__((ext_vector_type(8))) int          v8i;
typedef __attribute__((ext_vector_type(4))) int          v4i;

// ---------------------------------------------------------------------------
// Degree / normalization kernels
// ---------------------------------------------------------------------------
__global__ void deg_init_kernel(float* __restrict__ deg, int n) {
    int i = blockIdx.x * blockDim.x + threadIdx.x;
    if (i < n) deg[i] = 1.0f;                    // self-loop contribution
}

__global__ void deg_accum_kernel(const int* __restrict__ dst,
                                 float* __restrict__ deg, int e) {
    int i = blockIdx.x * blockDim.x + threadIdx.x;
    if (i < e) atomicAdd(&deg[dst[i]], 1.0f);
}

__global__ void deg_to_dinv_kernel(float* __restrict__ deg, int n) {
    int i = blockIdx.x * blockDim.x + threadIdx.x;
    if (i < n) deg[i] = rsqrtf(deg[i]);          // deg >= 1 always
}

// ---------------------------------------------------------------------------
// WMMA f32 GEMM:  Y[nrows x NCOLS] = X[nrows x 128] @ W[128 x NCOLS]
//  - W (128*NCOLS*4B <= 64 KB) staged in LDS via the Tensor Data Mover
//    (single tensor_load_to_lds issued by wave 0, s_wait_tensorcnt, barrier),
//    with a cooperative-load fallback when the TDM builtin is unavailable.
//  - optional fused bias + relu on store
//  - optional fused self-loop init: AGG[row][col] = Y[row][col] * dinv[row]^2
// One block = 8 waves = 128 rows; each wave owns a 16-row strip and
// NCOLS/16 accumulator tiles, K-loop of 32 x V_WMMA_F32_16X16X4_F32 steps.
// A fragment layout (32-bit 16x4): v0 = {K+0 | K+2}, v1 = {K+1 | K+3},
// rows m0..m0+15 across lanes (lane&15) in both halves.
// ---------------------------------------------------------------------------
template <int NCOLS>
__global__ __launch_bounds__(256) void gemm_wmma_f32_kernel(
    const float* __restrict__ X, const float* __restrict__ W,
    const float* __restrict__ bias, float* __restrict__ Y,
    const float* __restrict__ dinv, float* __restrict__ AGG,
    int nrows, int relu)
{
    constexpr int K  = 128;
    constexpr int NT = NCOLS / 16;

    __shared__ float Wlds[K * NCOLS];

#if defined(__gfx1250__) && __has_builtin(__builtin_amdgcn_tensor_load_to_lds)
    // ---- TDM: one DMA descriptor moves the whole 128 x NCOLS f32 tile ----
    if ((threadIdx.x >> 5) == 0) {               // wave 0 only (EXEC ignored by TDM)
        const unsigned long long ga = (unsigned long long)(size_t)W;
        const unsigned int lds = (unsigned int)(size_t)&Wlds[0]; // flat[31:0] == LDS addr
        // D# group0: count=1 | lds_addr | global_addr[56:0] | type=2
        v4u g0 = { 1u,
                   lds,
                   (unsigned int)ga,
                   ((unsigned int)((ga >> 32) & 0x01FFFFFFull)) | 0x80000000u };
        // D# group1: data_size=4B (code 2) @bits17:16; tensor_dim0=NCOLS @79:48;
        // tensor_dim1=128 @111:80; tile_dim0=NCOLS @127:112; tile_dim1=128 @143:128;
        // tensor_dim0_stride=NCOLS @207:160
        v8i g1 = { (int)0x00020000,
                   (int)((NCOLS & 0xFFFF) << 16),
                   (int)(128 << 16),
                   (int)(NCOLS << 16),
                   (int)128,
                   (int)NCOLS,
                   0, 0 };
        v4i gz4 = { 0, 0, 0, 0 };
#if defined(__clang_major__) && (__clang_major__ >= 23)
        v8i gz8 = { 0, 0, 0, 0, 0, 0, 0, 0 };
        __builtin_amdgcn_tensor_load_to_lds(g0, g1, gz4, gz4, gz8, 0);
#else
        __builtin_amdgcn_tensor_load_to_lds(g0, g1, gz4, gz4, 0);
#endif
        __builtin_amdgcn_s_wait_tensorcnt(0);
    }
    __syncthreads();
#else
    for (int i = threadIdx.x; i < (K * NCOLS) / 4; i += blockDim.x)
        ((float4*)Wlds)[i] = ((const float4*)W)[i];
    __syncthreads();
#endif

    const int wave    = threadIdx.x >> 5;    // 0..7
    const int lane    = threadIdx.x & 31;
    const int l16     = lane & 15;
    const int halfsel = lane >> 4;           // 0: K+0/K+1  1: K+2/K+3
    const int m0      = blockIdx.x * 128 + wave * 16;

    v8f acc[NT] = {};

    int mr = m0 + l16;
    if (mr >= nrows) mr = nrows - 1;         // clamp (stores are guarded)
    const float* xrow = X + (size_t)mr * K;

    for (int k = 0; k < K; k += 4) {
        const int kb = k + (halfsel << 1);
        v2f a;
        a.x = xrow[kb];
        a.y = xrow[kb + 1];
#pragma unroll
        for (int t = 0; t < NT; ++t) {
            const int col = t * 16 + l16;
            v2f b;
            b.x = Wlds[kb * NCOLS + col];
            b.y = Wlds[(kb + 1) * NCOLS + col];
            acc[t] = __builtin_amdgcn_wmma_f32_16x16x4_f32(
                /*neg_a=*/false, a, /*neg_b=*/false, b,
                /*c_mod=*/(short)0, acc[t],
                /*reuse_a=*/false, /*reuse_b=*/false);
        }
    }

#pragma unroll
    for (int i = 0; i < 8; ++i) {
        const int row = m0 + i + halfsel * 8;   // C/D layout: half rows per lane group
        if (row >= nrows) continue;
        float dsq = 0.0f;
        if (AGG) { const float di = dinv[row]; dsq = di * di; }
#pragma unroll
        for (int t = 0; t < NT; ++t) {
            const int col = t * 16 + l16;
            float v = acc[t][i];
            if (bias) v += bias[col];
            if (relu) v = fmaxf(v, 0.0f);
            Y[(size_t)row * NCOLS + col] = v;
            if (AGG) AGG[(size_t)row * NCOLS + col] = v * dsq;
        }
    }
}

// ---------------------------------------------------------------------------
// One wave per edge: lanes move float4 (32*16B = 512B row), scatter with
// global f32 atomics: AGG[dst] += H[src] * dinv[src]*dinv[dst]
// ---------------------------------------------------------------------------
__global__ __launch_bounds__(256) void edge_scatter_kernel(
    const float* __restrict__ H, const int* __restrict__ src,
    const int* __restrict__ dst, const float* __restrict__ dinv,
    float* __restrict__ AGG, int nedges)
{
    const int warp = blockIdx.x * (blockDim.x >> 5) + (threadIdx.x >> 5);
    const int lane = threadIdx.x & 31;
    if (warp >= nedges) return;
    const int s = src[warp];
    const int d = dst[warp];
    const float nrm = dinv[s] * dinv[d];

    const float4 v = ((const float4*)(H + (size_t)s * 128))[lane];
    float* ad = AGG + (size_t)d * 128 + lane * 4;
    atomicAdd(ad + 0, v.x * nrm);
    atomicAdd(ad + 1, v.y * nrm);
    atomicAdd(ad + 2, v.z * nrm);
    atomicAdd(ad + 3, v.w * nrm);
}

// ---------------------------------------------------------------------------
// In-place bias + relu over N x 128
// ---------------------------------------------------------------------------
__global__ void bias_relu_kernel(float* __restrict__ A,
                                 const float* __restrict__ b, size_t total) {
    size_t i = (size_t)blockIdx.x * blockDim.x + threadIdx.x;
    if (i >= total) return;
    A[i] = fmaxf(A[i] + b[i & 127], 0.0f);
}

// ---------------------------------------------------------------------------
// Final tiny projections: out[n] = { P1[n]@Wp2 + bp2 (2), T1[n]@Wt2 + bt2 (1) }
// One wave per node, cross-lane reduce via shfl_xor (wave32).
// ---------------------------------------------------------------------------
__global__ __launch_bounds__(256) void heads_final_kernel(
    const float* __restrict__ P1, const float* __restrict__ T1,
    const float* __restrict__ Wp2, const float* __restrict__ bp2,
    const float* __restrict__ Wt2, const float* __restrict__ bt2,
    float* __restrict__ out, int n)
{
    const int node = blockIdx.x * (blockDim.x >> 5) + (threadIdx.x >> 5);
    const int lane = threadIdx.x & 31;
    if (node >= n) return;
    const float* p = P1 + (size_t)node * 128;
    const float* t = T1 + (size_t)node * 64;

    float s0 = 0.0f, s1 = 0.0f, s2 = 0.0f;
#pragma unroll
    for (int j = lane; j < 128; j += 32) {
        const float pv = p[j];
        s0 += pv * Wp2[j * 2 + 0];
        s1 += pv * Wp2[j * 2 + 1];
    }
#pragma unroll
    for (int j = lane; j < 64; j += 32)
        s2 += t[j] * Wt2[j];

    for (int off = 16; off > 0; off >>= 1) {
        s0 += __shfl_xor(s0, off, 32);
        s1 += __shfl_xor(s1, off, 32);
        s2 += __shfl_xor(s2, off, 32);
    }
    if (lane == 0) {
        out[(size_t)node * 3 + 0] = s0 + bp2[0];
        out[(size_t)node * 3 + 1] = s1 + bp2[1];
        out[(size_t)node * 3 + 2] = s2 + bt2[0];
    }
}

// ---------------------------------------------------------------------------
extern "C" void kernel_launch(void* const* d_in, const int* in_sizes, int n_in,
                              void* d_out, int out_size, void* d_ws, size_t ws_size,
                              hipStream_t stream) {
    const float* x   = (const float*)d_in[0];
    const int*   ei  = (const int*)d_in[1];
    const float* W1  = (const float*)d_in[2];
    const float* b1  = (const float*)d_in[3];
    const float* W2  = (const float*)d_in[4];
    const float* b2  = (const float*)d_in[5];
    const float* W3  = (const float*)d_in[6];
    const float* b3  = (const float*)d_in[7];
    const float* Wp1 = (const float*)d_in[8];
    const float* bp1 = (const float*)d_in[9];
    const float* Wp2 = (const float*)d_in[10];
    const float* bp2 = (const float*)d_in[11];
    const float* Wt1 = (const float*)d_in[12];
    const float* bt1 = (const float*)d_in[13];
    const float* Wt2 = (const float*)d_in[14];
    const float* bt2 = (const float*)d_in[15];

    const int N = in_sizes[0] / 128;
    const int E = in_sizes[1] / 2;
    const int* src = ei;
    const int* dst = ei + E;

    // workspace layout: buf0 [N*128], buf1 [N*128], bufT [N*64], dinv [N]
    float* buf0 = (float*)d_ws;
    float* buf1 = buf0 + (size_t)N * 128;
    float* bufT = buf1 + (size_t)N * 128;
    float* dinv = bufT + (size_t)N * 64;

    const size_t totalNH = (size_t)N * 128;
    const int nodeBlocks = (N + 255) / 256;
    const int edgeBlocksT = (E + 255) / 256;
    const int elemBlocks = (int)((totalNH + 255) / 256);
    const int gemmBlocks = (N + 127) / 128;
    const int edgeBlocks = (E + 7) / 8;        // 8 waves per block, wave/edge
    const int waveNodeBlocks = (N + 7) / 8;

    // --- degree / normalization ---
    deg_init_kernel<<<nodeBlocks, 256, 0, stream>>>(dinv, N);
    deg_accum_kernel<<<edgeBlocksT, 256, 0, stream>>>(dst, dinv, E);
    deg_to_dinv_kernel<<<nodeBlocks, 256, 0, stream>>>(dinv, N);

    // --- 3 GCN conv layers ---
    // gemm writes H -> buf0 and the self-loop-initialized accumulator
    // AGG = H*dinv^2 -> buf1 in one pass; scatter adds edge messages into
    // buf1; bias+relu finishes buf1 in place; buf1 feeds the next layer.
    const float* xin = x;
    const float* Ws[3] = {W1, W2, W3};
    const float* bs[3] = {b1, b2, b3};
    for (int l = 0; l < 3; ++l) {
        gemm_wmma_f32_kernel<128><<<gemmBlocks, 256, 0, stream>>>(
            xin, Ws[l], nullptr, buf0, dinv, buf1, N, 0);
        edge_scatter_kernel<<<edgeBlocks, 256, 0, stream>>>(
            buf0, src, dst, dinv, buf1, E);
        bias_relu_kernel<<<elemBlocks, 256, 0, stream>>>(buf1, bs[l], totalNH);
        xin = buf1;
    }

    // --- heads ---
    gemm_wmma_f32_kernel<128><<<gemmBlocks, 256, 0, stream>>>(
        buf1, Wp1, bp1, buf0, nullptr, nullptr, N, 1);   // P1 -> buf0
    gemm_wmma_f32_kernel<64><<<gemmBlocks, 256, 0, stream>>>(
        buf1, Wt1, bt1, bufT, nullptr, nullptr, N, 1);   // T1 -> bufT
    heads_final_kernel<<<waveNodeBlocks, 256, 0, stream>>>(
        buf0, bufT, Wp2, bp2, Wt2, bt2, (float*)d_out, N);
}